// SelfAttention_45380624449736
// MI455X (gfx1250) — compile-verified
//
#include <hip/hip_runtime.h>
#include <hip/hip_bf16.h>

typedef __attribute__((ext_vector_type(2))) float v2f;
typedef __attribute__((ext_vector_type(8))) float v8f;

#define WMMA_F32X4(a, b, c) \
    __builtin_amdgcn_wmma_f32_16x16x4_f32(false, (a), false, (b), (short)0, (c), false, false)

namespace {
constexpr int B  = 4;
constexpr int C  = 256;
constexpr int CQ = 32;
constexpr int N  = 4096;              // W*H = 64*64
constexpr int LDS_STRIDE = N + 4;     // pad: avoids 16-way bank conflicts on transposed reads
}

// ---------------------------------------------------------------------------
// Pass 1: channel projection  out[b][r][n] = sum_c W[r][c] * x[b][c][n] + bias[r]
// Layouts: x (B,C,N) row-major, W (R,C) row-major, out (B,R,N) row-major.
// Block = 256 threads = 8 waves; each wave owns one 16x16 output tile, K=C=256.
// Dual accumulator chains break the WMMA C->D serial dependency.
// ---------------------------------------------------------------------------
__global__ __launch_bounds__(256) void proj_wmma_kernel(
    const float* __restrict__ x, const float* __restrict__ W,
    const float* __restrict__ bias, float* __restrict__ out, int R)
{
    const int lane = threadIdx.x & 31;
    const int wave = threadIdx.x >> 5;
    const int b    = blockIdx.z;
    const int r0   = blockIdx.y * 16;
    const int n0   = (blockIdx.x * 8 + wave) * 16;
    const int c16  = lane & 15;
    const int half = (lane < 16) ? 0 : 2;   // K split across lane halves (ISA 7.12.2)

    const float* xb = x + (size_t)b * C * N;
    const float* wr = W + (size_t)(r0 + c16) * C;

    v8f acc0 = {}, acc1 = {};
    for (int kk = 0; kk < C / 4; kk += 2) {
        const int kb0 = 4 * kk + half;
        const int kb1 = kb0 + 4;
        v2f a0, b0, a1, b1;
        a0.x = wr[kb0];     a0.y = wr[kb0 + 1];
        a1.x = wr[kb1];     a1.y = wr[kb1 + 1];
        b0.x = xb[(size_t)kb0 * N + n0 + c16];
        b0.y = xb[(size_t)(kb0 + 1) * N + n0 + c16];
        b1.x = xb[(size_t)kb1 * N + n0 + c16];
        b1.y = xb[(size_t)(kb1 + 1) * N + n0 + c16];
        acc0 = WMMA_F32X4(a0, b0, acc0);
        acc1 = WMMA_F32X4(a1, b1, acc1);
    }
    const v8f acc = acc0 + acc1;

    // D layout: VGPR r -> row r (lanes 0-15) / r+8 (lanes 16-31), col = lane%16
    const int rbase = r0 + ((lane < 16) ? 0 : 8);
    float* ob = out + (size_t)b * R * N;
#pragma unroll
    for (int r = 0; r < 8; ++r) {
        const int row = rbase + r;
        ob[(size_t)row * N + n0 + c16] = acc[r] + bias[row];
    }
}

// ---------------------------------------------------------------------------
// Pass 2: fused  S = qKᵀ -> softmax -> write attention -> out = V·Pᵀ, + gamma*out + x
// One workgroup (512 thr = 16 waves) per (batch, 16-query-row tile).
// Whole 16x4096 score block lives in 256.25 KB of the 320 KB LDS.
// ---------------------------------------------------------------------------
__global__ __launch_bounds__(512) void attn_fused_kernel(
    const float* __restrict__ qT, const float* __restrict__ kT,
    const float* __restrict__ vT, const float* __restrict__ x,
    const float* __restrict__ gamma, float* __restrict__ outAll)
{
    extern __shared__ float S[];          // [16][LDS_STRIDE]
    const int lane = threadIdx.x & 31;
    const int wave = threadIdx.x >> 5;    // 0..15
    const int b    = blockIdx.y;
    const int m0   = blockIdx.x * 16;     // query-row tile
    const int c16  = lane & 15;
    const int half = (lane < 16) ? 0 : 2;

    const float* qb = qT + (size_t)b * CQ * N;
    const float* kb = kT + (size_t)b * CQ * N;

    // Preload A fragments: q rows m0..m0+15, K=32 -> 8 fragments (reused 16x)
    v2f aq[8];
#pragma unroll
    for (int kk = 0; kk < 8; ++kk) {
        const int r = 4 * kk + half;
        aq[kk].x = qb[(size_t)r * N + m0 + c16];
        aq[kk].y = qb[(size_t)(r + 1) * N + m0 + c16];
    }

    // ---- Stage 1: S[16][4096] = q @ k ; wave w covers n-tiles w, w+16, ...
    for (int t = wave; t < N / 16; t += 16) {
        const int n0 = t * 16;
        v8f acc0 = {}, acc1 = {};
#pragma unroll
        for (int kk = 0; kk < 8; kk += 2) {
            const int r0 = 4 * kk + half;
            const int r1 = r0 + 4;
            v2f b0, b1;
            b0.x = kb[(size_t)r0 * N + n0 + c16];
            b0.y = kb[(size_t)(r0 + 1) * N + n0 + c16];
            b1.x = kb[(size_t)r1 * N + n0 + c16];
            b1.y = kb[(size_t)(r1 + 1) * N + n0 + c16];
            acc0 = WMMA_F32X4(aq[kk],     b0, acc0);
            acc1 = WMMA_F32X4(aq[kk + 1], b1, acc1);
        }
        const v8f acc = acc0 + acc1;
        const int rowb = (lane < 16) ? 0 : 8;
#pragma unroll
        for (int r = 0; r < 8; ++r)
            S[(size_t)(rowb + r) * LDS_STRIDE + n0 + c16] = acc[r];
    }
    __syncthreads();

    // ---- Stage 2: softmax on row `wave` (one wave per query row), write attention once
    {
        float* row = S + (size_t)wave * LDS_STRIDE;     // 16B-aligned (stride*4 % 16 == 0)
        float mx = -INFINITY;
        for (int j = lane * 4; j < N; j += 128) {
            const float4 s = *(const float4*)(row + j);
            mx = fmaxf(mx, fmaxf(fmaxf(s.x, s.y), fmaxf(s.z, s.w)));
        }
#pragma unroll
        for (int off = 16; off > 0; off >>= 1) mx = fmaxf(mx, __shfl_xor(mx, off, 32));

        float sum = 0.0f;
        for (int j = lane * 4; j < N; j += 128) {
            float4 s = *(const float4*)(row + j);
            s.x = __expf(s.x - mx); s.y = __expf(s.y - mx);
            s.z = __expf(s.z - mx); s.w = __expf(s.w - mx);
            *(float4*)(row + j) = s;
            sum += s.x + s.y + s.z + s.w;
        }
#pragma unroll
        for (int off = 16; off > 0; off >>= 1) sum += __shfl_xor(sum, off, 32);
        const float scale = 1.0f / sum;

        float* attg = outAll + (size_t)B * C * N
                    + ((size_t)b * N + (m0 + wave)) * N;
        for (int j = lane * 4; j < N; j += 128) {
            float4 s = *(const float4*)(row + j);
            s.x *= scale; s.y *= scale; s.z *= scale; s.w *= scale;
            *(float4*)(row + j)  = s;        // normalized probs for stage 3
            *(float4*)(attg + j) = s;        // attention output (touched once in HBM)
        }
    }
    __syncthreads();

    // ---- Stage 3: out[c, m0..m0+15] = V(256x4096) @ Pᵀ ; wave w owns c-tile w, K=4096.
    //      Dual accumulator chains + 4x unroll: dominant 34-GFLOP loop.
    {
        const float* vb   = vT + (size_t)b * C * N;
        const int    c0   = wave * 16;
        const float* vrow = vb + (size_t)(c0 + c16) * N + half;      // A: v rows
        const float* prow = S + (size_t)c16 * LDS_STRIDE + half;     // B: probs (query=c16)
        v8f acc0 = {}, acc1 = {};
        for (int kk = 0; kk < N / 4; kk += 4) {
            const v2f a0 = *(const v2f*)(vrow + 4 * kk);
            const v2f p0 = *(const v2f*)(prow + 4 * kk);
            const v2f a1 = *(const v2f*)(vrow + 4 * kk + 4);
            const v2f p1 = *(const v2f*)(prow + 4 * kk + 4);
            const v2f a2 = *(const v2f*)(vrow + 4 * kk + 8);
            const v2f p2 = *(const v2f*)(prow + 4 * kk + 8);
            const v2f a3 = *(const v2f*)(vrow + 4 * kk + 12);
            const v2f p3 = *(const v2f*)(prow + 4 * kk + 12);
            acc0 = WMMA_F32X4(a0, p0, acc0);
            acc1 = WMMA_F32X4(a1, p1, acc1);
            acc0 = WMMA_F32X4(a2, p2, acc0);
            acc1 = WMMA_F32X4(a3, p3, acc1);
        }
        const v8f acc = acc0 + acc1;
        const float g  = gamma[0];
        const float* xb = x + (size_t)b * C * N;
        float* ob = outAll + (size_t)b * C * N;
        const int rowb = c0 + ((lane < 16) ? 0 : 8);
#pragma unroll
        for (int r = 0; r < 8; ++r) {
            const size_t idx = (size_t)(rowb + r) * N + m0 + c16;
            ob[idx] = g * acc[r] + xb[idx];
        }
    }
}

// ---------------------------------------------------------------------------
extern "C" void kernel_launch(void* const* d_in, const int* in_sizes, int n_in,
                              void* d_out, int out_size, void* d_ws, size_t ws_size,
                              hipStream_t stream)
{
    const float* x     = (const float*)d_in[0];
    const float* Wq    = (const float*)d_in[1];
    const float* bq    = (const float*)d_in[2];
    const float* Wk    = (const float*)d_in[3];
    const float* bk    = (const float*)d_in[4];
    const float* Wv    = (const float*)d_in[5];
    const float* bv    = (const float*)d_in[6];
    const float* gamma = (const float*)d_in[7];
    float* out = (float*)d_out;

    float* qT = (float*)d_ws;                    //  (B, 32, N)  2 MB
    float* kT = qT + (size_t)B * CQ * N;         //  (B, 32, N)  2 MB
    float* vT = kT + (size_t)B * CQ * N;         //  (B, 256, N) 16 MB

    // QKV projections (WMMA f32)
    proj_wmma_kernel<<<dim3(N / 128, CQ / 16, B), 256, 0, stream>>>(x, Wq, bq, qT, CQ);
    proj_wmma_kernel<<<dim3(N / 128, CQ / 16, B), 256, 0, stream>>>(x, Wk, bk, kT, CQ);
    proj_wmma_kernel<<<dim3(N / 128, C  / 16, B), 256, 0, stream>>>(x, Wv, bv, vT, C);

    // Fused scores -> softmax -> attention write -> out GEMM
    const size_t smem = (size_t)16 * LDS_STRIDE * sizeof(float);   // 256.25 KB < 320 KB
    attn_fused_kernel<<<dim3(N / 16, B), 512, smem, stream>>>(qT, kT, vT, x, gamma, out);
}